// BandwidthPredictorNNHall_41455024341448
// MI455X (gfx1250) — compile-verified
//
#include <hip/hip_runtime.h>
#include <math.h>

typedef __attribute__((ext_vector_type(2))) float v2f;
typedef __attribute__((ext_vector_type(8))) float v8f;

#define INV_SQRT_2PI 0.3989422804014327f
#define RK_CONST 0.282095f

// ---------------------------------------------------------------------------
// Kernel 1: per-batch std (ddof=1) -> pilot, zero accumulators, Q = P / pilot
// One block per batch, 256 threads.
// ---------------------------------------------------------------------------
__global__ void k_prep(const float* __restrict__ p, float* __restrict__ q,
                       float* __restrict__ pilot, float* __restrict__ s2,
                       float* __restrict__ s3, int n, float npow) {
  const int b = blockIdx.x;
  const int tid = threadIdx.x;
  const float4* pb = reinterpret_cast<const float4*>(p) + (size_t)b * n;
  float4* qb = reinterpret_cast<float4*>(q) + (size_t)b * n;

  float sum[4] = {0.f, 0.f, 0.f, 0.f};
  float sq[4]  = {0.f, 0.f, 0.f, 0.f};
  for (int i = tid; i < n; i += blockDim.x) {
    float4 v = pb[i];
    sum[0] += v.x; sum[1] += v.y; sum[2] += v.z; sum[3] += v.w;
    sq[0] += v.x * v.x; sq[1] += v.y * v.y;
    sq[2] += v.z * v.z; sq[3] += v.w * v.w;
  }

  __shared__ float red[8][256];
  for (int k = 0; k < 4; ++k) { red[k][tid] = sum[k]; red[k + 4][tid] = sq[k]; }
  __syncthreads();
  for (int s = 128; s > 0; s >>= 1) {
    if (tid < s) {
      for (int k = 0; k < 8; ++k) red[k][tid] += red[k][tid + s];
    }
    __syncthreads();
  }

  __shared__ float pl[4];
  if (tid < 4) {
    float s1 = red[tid][0];
    float ss = red[tid + 4][0];
    float var = (ss - s1 * s1 / (float)n) / (float)(n - 1);  // ddof = 1
    float pv = 1.0592f * sqrtf(var) * npow;
    pl[tid] = pv;
    pilot[b * 4 + tid] = pv;
    s2[b * 4 + tid] = 0.f;
    s3[b * 4 + tid] = 0.f;
  }
  __syncthreads();

  const float ix = 1.f / pl[0], iy = 1.f / pl[1], iz = 1.f / pl[2], iw = 1.f / pl[3];
  for (int i = tid; i < n; i += blockDim.x) {
    float4 v = pb[i];
    qb[i] = make_float4(v.x * ix, v.y * iy, v.z * iz, v.w * iw);
  }
}

// ---------------------------------------------------------------------------
// Kernel 2: pairwise accumulation. One wave32 per (batch, i_tile, j_chunk).
// Dot products Q_i . Q_j via V_WMMA_F32_16X16X4_F32 (A: 16x4, B: 4x16).
// kern = exp(dot - 0.5*r_i - 0.5*r_j); s2 += kern*(dx^2-1); s3 += kern*(dx^3-3dx)
// ---------------------------------------------------------------------------
__global__ void k_pair(const float* __restrict__ q, float* __restrict__ s2,
                       float* __restrict__ s3, int n, int numTiles,
                       int totalWaves) {
  const int gwave = (int)((blockIdx.x * blockDim.x + threadIdx.x) >> 5);
  if (gwave >= totalWaves) return;  // uniform per wave: EXEC stays all-ones
  const int lane = threadIdx.x & 31;
  const int perB = numTiles * 4;
  const int b = gwave / perB;
  const int rem = gwave - b * perB;
  const int i_tile = rem >> 2;
  const int jchunk = rem & 3;

  const float4* Qb = reinterpret_cast<const float4*>(q) + (size_t)b * n;
  const int i_base = i_tile * 16;
  const int half = lane >> 4;   // 0: lanes 0-15 carry K0,K1 | 1: lanes 16-31 carry K2,K3
  const int l15 = lane & 15;

  // A operand (16x4 f32): lane half selects K pair of row i_base + (lane&15)
  float4 qa = Qb[i_base + l15];
  v2f A;
  if (half) { A.x = qa.z; A.y = qa.w; } else { A.x = qa.x; A.y = qa.y; }

  // D-layout rows owned by this lane: M = v + 8*half  (v = accumulator vgpr 0..7)
  const int mbase = i_base + half * 8;
  float4 qi[8];
  float ri[8];
#pragma unroll
  for (int v = 0; v < 8; ++v) {
    qi[v] = Qb[mbase + v];
    ri[v] = qi[v].x * qi[v].x + qi[v].y * qi[v].y +
            qi[v].z * qi[v].z + qi[v].w * qi[v].w;
  }

  float acc2[4] = {0.f, 0.f, 0.f, 0.f};
  float acc3[4] = {0.f, 0.f, 0.f, 0.f};

  for (int jt = jchunk; jt < numTiles; jt += 4) {
    const int j_base = jt * 16;
    // B operand (4x16 f32), mirror of A layout; D column for this lane is
    // N = lane&15, so the same row feeds both B and the per-column q_j vector.
    float4 qj = Qb[j_base + l15];
    v2f B;
    if (half) { B.x = qj.z; B.y = qj.w; } else { B.x = qj.x; B.y = qj.y; }
    const float rj = qj.x * qj.x + qj.y * qj.y + qj.z * qj.z + qj.w * qj.w;

    v8f c;
#pragma unroll
    for (int v = 0; v < 8; ++v) c[v] = -0.5f * rj;

    // D[m][n] = Q_i[m] . Q_j[n] - 0.5*r_j[n]
    c = __builtin_amdgcn_wmma_f32_16x16x4_f32(
        /*neg_a=*/false, A, /*neg_b=*/false, B,
        /*c_mod=*/(short)0, c, /*reuse_a=*/false, /*reuse_b=*/false);

#pragma unroll
    for (int v = 0; v < 8; ++v) {
      const float kern = __expf(c[v] - 0.5f * ri[v]);
      float dxa[4] = {qj.x - qi[v].x, qj.y - qi[v].y,
                      qj.z - qi[v].z, qj.w - qi[v].w};
#pragma unroll
      for (int dd = 0; dd < 4; ++dd) {
        const float dx = dxa[dd];
        const float dx2 = dx * dx;
        acc2[dd] = fmaf(kern, dx2 - 1.f, acc2[dd]);
        acc3[dd] = fmaf(kern * dx, dx2 - 3.f, acc3[dd]);
      }
    }
  }

  // Wave32 reduction
#pragma unroll
  for (int off = 16; off > 0; off >>= 1) {
#pragma unroll
    for (int dd = 0; dd < 4; ++dd) {
      acc2[dd] += __shfl_down(acc2[dd], off, 32);
      acc3[dd] += __shfl_down(acc3[dd], off, 32);
    }
  }
  if (lane == 0) {
#pragma unroll
    for (int dd = 0; dd < 4; ++dd) {
      atomicAdd(&s2[b * 4 + dd], acc2[dd]);
      atomicAdd(&s3[b * 4 + dd], acc3[dd]);
    }
  }
}

// ---------------------------------------------------------------------------
// Kernel 3: scalar tail -> bandwidth1 + bandwidth2, 32 outputs
// ---------------------------------------------------------------------------
__global__ void k_final(const float* __restrict__ pilot,
                        const float* __restrict__ s2,
                        const float* __restrict__ s3,
                        float* __restrict__ out, int n, int total) {
  const int t = blockIdx.x * blockDim.x + threadIdx.x;
  if (t >= total) return;
  const float pv = pilot[t];
  const float denom = (float)n * (float)(n - 1);
  const float p2 = pv * pv;
  const float p5 = p2 * p2 * pv;
  const float p7 = p5 * p2;
  const float I2 = (s2[t] * INV_SQRT_2PI) / (p5 * denom);
  const float I3 = -(s3[t] * INV_SQRT_2PI) / (p7 * denom);
  const float J1 = RK_CONST / I2;                 // MU2 = 1
  const float J2 = 3.0f * I3 / (20.0f * I2);      // MU4 = 3
  const float base = J1 / (float)n;
  const float sg = (base > 0.f) ? 1.f : ((base < 0.f) ? -1.f : 0.f);
  const float ab = fabsf(base);
  const float bw1 = sg * powf(ab, 0.2f);
  const float bw2 = sg * powf(ab, 0.6f) * J2;
  out[t] = bw1 + bw2;
}

// ---------------------------------------------------------------------------
extern "C" void kernel_launch(void* const* d_in, const int* in_sizes, int n_in,
                              void* d_out, int out_size, void* d_ws, size_t ws_size,
                              hipStream_t stream) {
  const float* particles = (const float*)d_in[0];
  // d_in[1] (weights) is unused by the reference.
  const int d = in_sizes[0] / in_sizes[1];   // 4
  const int b = out_size / d;                // 8
  const int n = in_sizes[1] / b;             // 1024

  float* ws = (float*)d_ws;
  float* q = ws;                                   // b*n*4 floats
  float* pilot = ws + (size_t)b * n * 4;           // b*4
  float* s2 = pilot + b * 4;                       // b*4
  float* s3 = s2 + b * 4;                          // b*4

  const float npow = powf((float)n, -1.0f / (4.0f + (float)d));

  k_prep<<<b, 256, 0, stream>>>(particles, q, pilot, s2, s3, n, npow);

  const int numTiles = n / 16;                     // 64
  const int totalWaves = b * numTiles * 4;         // 2048 waves
  const int threads = 128;                         // 4 waves per block
  const int blocks = (totalWaves * 32 + threads - 1) / threads;
  k_pair<<<blocks, threads, 0, stream>>>(q, s2, s3, n, numTiles, totalWaves);

  k_final<<<1, 64, 0, stream>>>(pilot, s2, s3, (float*)d_out, n, b * d);
}